// DenseGrid_2877628088738
// MI455X (gfx1250) — compile-verified
//
#include <hip/hip_runtime.h>
#include <cmath>

#define FEAT4 12            // 48 floats = 12 float4 per codebook row
#define BLOCK 256
#define WAVES_PER_BLOCK 8   // wave32: 256 threads = 8 waves

typedef __attribute__((ext_vector_type(4))) float v4f;

// CDNA5 async global->LDS staging (gfx1250). SADDR form:
//   global_load_async_to_lds_b64 vdst_lds_off, vaddr_off32, s[base:base+1]
// Tracked by ASYNCcnt; completion enforced with s_wait_asynccnt.
__device__ __forceinline__ void async_stage_pts(const float* pts,
                                                unsigned gByteOff,
                                                unsigned ldsByteOff) {
    asm volatile("global_load_async_to_lds_b64 %0, %1, %2"
                 :
                 : "v"(ldsByteOff), "v"(gByteOff), "s"(pts)
                 : "memory");
}

__global__ __launch_bounds__(BLOCK)
void bilerp48_kernel(const float* __restrict__ pts,
                     const v4f* __restrict__ cb,
                     v4f* __restrict__ out,
                     int n, int res)
{
    // [buffer][wave][point-slot] : 2 * 8 * 32 * 8B = 4 KB LDS
    __shared__ float2 sbuf[2][WAVES_PER_BLOCK][32];

    const int tid  = threadIdx.x;
    const int wave = tid >> 5;
    const int lane = tid & 31;
    const int quad = lane >> 2;   // 8 points per wave-pass
    const int chnk = lane & 3;    // which float4 column group

    const unsigned nTiles     = ((unsigned)n + 31u) >> 5;
    const unsigned totalWaves = (gridDim.x * (unsigned)blockDim.x) >> 5;
    unsigned tile = ((blockIdx.x * (unsigned)blockDim.x) + (unsigned)tid) >> 5;
    if (tile >= nTiles) return;   // wave-uniform

    // Low 32 bits of the flat shared-memory address == LDS segment offset
    // (ISA 10.2: LDS aperture).
    const unsigned lds0 = (unsigned)(size_t)&sbuf[0][wave][lane];
    const unsigned lds1 = (unsigned)(size_t)&sbuf[1][wave][lane];

    const float rscale = (float)(res - 1);
    const float rmax   = (float)(res - 2);

    // Prologue: stage first tile (32 points, 8B each) into buffer 0.
    {
        unsigned p = tile * 32u + (unsigned)lane;
        if (p >= (unsigned)n) p = (unsigned)n - 1u;   // clamp tail lanes
        async_stage_pts(pts, p * 8u, lds0);
    }

    int cur = 0;
    while (tile < nTiles) {
        // Prefetch next tile into the other buffer. Always issue exactly one
        // async op per iteration (clamped re-load on the tail) so the
        // ASYNCcnt discipline below stays uniform.
        const unsigned nextTile = tile + totalWaves;
        {
            const unsigned pfTile = (nextTile < nTiles) ? nextTile : tile;
            unsigned p = pfTile * 32u + (unsigned)lane;
            if (p >= (unsigned)n) p = (unsigned)n - 1u;
            async_stage_pts(pts, p * 8u, cur ? lds0 : lds1);
        }
        // Two async ops outstanding; wait until only the prefetch remains ->
        // the current tile's data is resident in LDS.
        asm volatile("s_wait_asynccnt 0x1" ::: "memory");

        #pragma unroll
        for (int s = 0; s < 4; ++s) {
            const int q = s * 8 + quad;
            const float2 p = sbuf[cur][wave][q];          // ds_load_b64 (quad broadcast)
            const unsigned pidx = tile * 32u + (unsigned)q;

            const float xs = p.x * rscale;
            const float ys = p.y * rscale;
            const float fx = fminf(fmaxf(floorf(xs), 0.f), rmax);
            const float fy = fminf(fmaxf(floorf(ys), 0.f), rmax);
            const float wx = xs - fx;
            const float wy = ys - fy;
            const int   ib = (int)fx * res + (int)fy;

            const v4f* r0 = cb + (size_t)ib * FEAT4;              // f00 row
            const v4f* r1 = r0 + (size_t)res * FEAT4;             // f10 row
            const float w00 = (1.f - wx) * (1.f - wy);
            const float w01 = (1.f - wx) * wy;
            const float w10 = wx * (1.f - wy);
            const float w11 = wx * wy;

            const bool ok = pidx < (unsigned)n;
            v4f* op = out + (size_t)pidx * FEAT4;

            #pragma unroll
            for (int k = 0; k < 3; ++k) {
                const int j = chnk + 4 * k;                        // quad covers 64B
                const v4f a = r0[j];                               // f00 (temporal: L2 reuse)
                const v4f b = r0[j + FEAT4];                       // f01 (row ib+1)
                const v4f d = r1[j];                               // f10 (row ib+res)
                const v4f e = r1[j + FEAT4];                       // f11
                v4f o;
                o.x = fmaf(e.x, w11, fmaf(d.x, w10, fmaf(b.x, w01, a.x * w00)));
                o.y = fmaf(e.y, w11, fmaf(d.y, w10, fmaf(b.y, w01, a.y * w00)));
                o.z = fmaf(e.z, w11, fmaf(d.z, w10, fmaf(b.z, w01, a.z * w00)));
                o.w = fmaf(e.w, w11, fmaf(d.w, w10, fmaf(b.w, w01, a.w * w00)));
                if (ok) {
                    // Streaming, write-once output: non-temporal store keeps
                    // the 384 MB result stream from evicting the L2-resident
                    // codebook (global_store_b128 th:TH_STORE_NT).
                    __builtin_nontemporal_store(o, op + j);
                }
            }
        }
        tile = nextTile;
        cur ^= 1;
    }
}

extern "C" void kernel_launch(void* const* d_in, const int* in_sizes, int n_in,
                              void* d_out, int out_size, void* d_ws, size_t ws_size,
                              hipStream_t stream) {
    (void)n_in; (void)out_size; (void)d_ws; (void)ws_size;
    const float* pts = (const float*)d_in[0];
    const v4f*   cb  = (const v4f*)  d_in[1];
    v4f*         out = (v4f*)d_out;

    const int n    = in_sizes[0] / 2;              // 2 floats per point
    const int rows = in_sizes[1] / 48;             // FEAT_DIM = 48
    const int res  = (int)(sqrt((double)rows) + 0.5);
    if (n <= 0) return;

    const unsigned nTiles = ((unsigned)n + 31u) >> 5;
    unsigned blocks = (nTiles + WAVES_PER_BLOCK - 1) / WAVES_PER_BLOCK;
    if (blocks > 2048u) blocks = 2048u;            // grid-stride, ~4 tiles/wave

    hipLaunchKernelGGL(bilerp48_kernel, dim3(blocks), dim3(BLOCK), 0, stream,
                       pts, cb, out, n, res);
}